// GCNConvLayer_20590073217125
// MI455X (gfx1250) — compile-verified
//
#include <hip/hip_runtime.h>
#include <hip/hip_bf16.h>

typedef __attribute__((ext_vector_type(2))) float v2f;
typedef __attribute__((ext_vector_type(8))) float v8f;

#define D 128
#define WAVE 32

// ---------------------------------------------------------------------------
// degree / normalization
// ---------------------------------------------------------------------------
__global__ void deg_init_kernel(float* deg, int N) {
    int i = blockIdx.x * blockDim.x + threadIdx.x;
    if (i < N) deg[i] = 1.0f;   // self loop
}

__global__ void deg_acc_kernel(const int* dst, float* deg, int E) {
    int i = blockIdx.x * blockDim.x + threadIdx.x;
    if (i < E) atomicAdd(&deg[dst[i]], 1.0f);
}

__global__ void deg_inv_kernel(float* deg, int N) {
    int i = blockIdx.x * blockDim.x + threadIdx.x;
    if (i < N) {
        float v = deg[i];
        deg[i] = v > 0.0f ? rsqrtf(v) : 0.0f;
    }
}

// ---------------------------------------------------------------------------
// fp32 WMMA GEMM: C[M x Nc] = A[M x K] @ B[K x Nc],  M % 32 == 0
// one wave computes a 32x16 strip (two 16x16 tiles sharing the B fragment)
// ---------------------------------------------------------------------------
__global__ __launch_bounds__(256) void gemm_f32_kernel(
    const float* __restrict__ A, const float* __restrict__ B,
    float* __restrict__ C, int M, int K, int Nc) {
    int tiles_n = Nc >> 4;
    int total   = (M >> 5) * tiles_n;          // 32-row strips
    int t = blockIdx.x * 8 + (threadIdx.x >> 5);
    if (t >= total) return;                    // wave-uniform
    int row0 = (t / tiles_n) << 5;
    int col0 = (t % tiles_n) << 4;
    int lane = threadIdx.x & 31;
    int l  = lane & 15;
    int hi = lane >> 4;                        // 0: K=0,1  1: K=2,3

    v8f c0 = {}, c1 = {};
    const float* a0 = A + (size_t)(row0 + l) * K;
    const float* a1 = A + (size_t)(row0 + 16 + l) * K;
    for (int k0 = 0; k0 < K; k0 += 4) {
        v2f b;
        b.x = B[(size_t)(k0 + 2 * hi) * Nc + col0 + l];
        b.y = B[(size_t)(k0 + 2 * hi + 1) * Nc + col0 + l];
        v2f av0 = *(const v2f*)(a0 + k0 + 2 * hi);
        v2f av1 = *(const v2f*)(a1 + k0 + 2 * hi);
        c0 = __builtin_amdgcn_wmma_f32_16x16x4_f32(false, av0, false, b,
                                                   (short)0, c0, false, false);
        c1 = __builtin_amdgcn_wmma_f32_16x16x4_f32(false, av1, false, b,
                                                   (short)0, c1, false, false);
    }
    for (int r = 0; r < 8; ++r) {
        int m = r + hi * 8;
        C[(size_t)(row0 + m) * Nc + col0 + l]      = c0[r];
        C[(size_t)(row0 + 16 + m) * Nc + col0 + l] = c1[r];
    }
}

// ---------------------------------------------------------------------------
// agg init with GCN bias
// ---------------------------------------------------------------------------
__global__ void agg_init_kernel(float* agg, const float* b, size_t total) {
    for (size_t i = blockIdx.x * (size_t)blockDim.x + threadIdx.x; i < total;
         i += (size_t)gridDim.x * blockDim.x)
        agg[i] = b[i & (D - 1)];
}

// ---------------------------------------------------------------------------
// edge scatter: one wave per (edge | self loop); lane covers 4 columns
// agg[dst] += hW[src] * dinv[src]*dinv[dst]   (L2-resident gather + atomics)
// ---------------------------------------------------------------------------
__global__ __launch_bounds__(256) void edge_scatter_kernel(
    const int* __restrict__ ei, const float* __restrict__ hW,
    const float* __restrict__ dinv, float* __restrict__ agg, int E, int N) {
    long long wid = ((long long)blockIdx.x * blockDim.x + threadIdx.x) >> 5;
    int lane = threadIdx.x & 31;
    if (wid >= (long long)E + N) return;  // wave-uniform
    int s, d;
    if (wid < E) { s = ei[wid]; d = ei[(size_t)E + wid]; }
    else         { s = d = (int)(wid - E); }
    float nrm = dinv[s] * dinv[d];
    float4 v = ((const float4*)(hW + (size_t)s * D))[lane];
    float* dst = agg + (size_t)d * D + lane * 4;
    atomicAdd(dst + 0, v.x * nrm);
    atomicAdd(dst + 1, v.y * nrm);
    atomicAdd(dst + 2, v.z * nrm);
    atomicAdd(dst + 3, v.w * nrm);
}

// ---------------------------------------------------------------------------
// batchnorm column statistics (two-pass)
// ---------------------------------------------------------------------------
__global__ void stats_zero_kernel(float* sum) {   // sum[128] then sumsq[128]
    sum[threadIdx.x] = 0.0f;                      // 256 threads
}

__global__ __launch_bounds__(256) void colstats_kernel(
    const float* __restrict__ src, float* __restrict__ sum,
    float* __restrict__ sumsq, int N) {
    int col  = threadIdx.x & (D - 1);
    int half = threadIdx.x >> 7;                  // 0 or 1
    float s = 0.0f, q = 0.0f;
    for (int r = blockIdx.x * 2 + half; r < N; r += gridDim.x * 2) {
        float v = src[(size_t)r * D + col];
        s += v; q += v * v;
    }
    __shared__ float ls[256], lq[256];
    ls[threadIdx.x] = s; lq[threadIdx.x] = q;
    __syncthreads();
    if (half == 0) {
        atomicAdd(&sum[col],   ls[col] + ls[col + D]);
        atomicAdd(&sumsq[col], lq[col] + lq[col + D]);
    }
}

__global__ void bn_finalize_kernel(const float* sum, const float* sumsq,
                                   const float* gamma, const float* beta,
                                   float* scale, float* shift, float invN) {
    int c = threadIdx.x;                          // 128 threads
    float m   = sum[c] * invN;
    float var = sumsq[c] * invN - m * m;
    float inv = rsqrtf(var + 1e-5f);
    float sc  = gamma[c] * inv;
    scale[c] = sc;
    shift[c] = beta[c] - m * sc;
}

// out = (residual? residual[i] : 0) + act(scale[col]*src[i] + shift[col])
__global__ __launch_bounds__(256) void bn_apply_kernel(
    const float* __restrict__ src, const float* __restrict__ residual,
    const float* __restrict__ scale, const float* __restrict__ shift,
    float* __restrict__ out, size_t total, int do_relu) {
    for (size_t i = blockIdx.x * (size_t)blockDim.x + threadIdx.x; i < total;
         i += (size_t)gridDim.x * blockDim.x) {
        int col = (int)(i & (D - 1));
        float v = scale[col] * src[i] + shift[col];
        if (do_relu) v = v > 0.0f ? v : 0.0f;
        if (residual) v += residual[i];
        out[i] = v;
    }
}

// ---------------------------------------------------------------------------
// fused FFN per 32-row strip (M % 32 == 0):
//   ff1 = relu(h2 @ W1 + b1)   (32x256, kept in LDS)
//   out = h2 + ff1 @ W2 + b2   (32x128)
// 256 threads = 8 waves; each wave computes 32x16 strips (B-fragment reuse)
// LDS: 16KB (h2 strip) + 32KB (ff1 strip)
// ---------------------------------------------------------------------------
__global__ __launch_bounds__(256) void ffn_fused_kernel(
    const float* __restrict__ h2, const float* __restrict__ W1,
    const float* __restrict__ b1, const float* __restrict__ W2,
    const float* __restrict__ b2, float* __restrict__ out, int M) {
    __shared__ float sA[32 * 128];   // h2 strip
    __shared__ float sF[32 * 256];   // ff1 strip
    int row0 = blockIdx.x * 32;
    int wave = threadIdx.x >> 5;
    int lane = threadIdx.x & 31;
    int l  = lane & 15;
    int hi = lane >> 4;

    {   // stage strip to LDS (1024 float4)
        const float4* g  = (const float4*)(h2 + (size_t)row0 * 128);
        float4*       s4 = (float4*)sA;
        for (int i = threadIdx.x; i < 1024; i += 256) s4[i] = g[i];
    }
    __syncthreads();

    // phase 1: 16 column-tiles of ff1; each is a 32x16 strip
    for (int t = wave; t < 16; t += 8) {
        int col0 = t * 16;
        v8f c0 = {}, c1 = {};
        for (int k0 = 0; k0 < 128; k0 += 4) {
            v2f b;
            b.x = W1[(size_t)(k0 + 2 * hi) * 256 + col0 + l];
            b.y = W1[(size_t)(k0 + 2 * hi + 1) * 256 + col0 + l];
            v2f a0 = *(const v2f*)(sA + l * 128 + k0 + 2 * hi);
            v2f a1 = *(const v2f*)(sA + (16 + l) * 128 + k0 + 2 * hi);
            c0 = __builtin_amdgcn_wmma_f32_16x16x4_f32(false, a0, false, b,
                                                       (short)0, c0, false, false);
            c1 = __builtin_amdgcn_wmma_f32_16x16x4_f32(false, a1, false, b,
                                                       (short)0, c1, false, false);
        }
        float bias = b1[col0 + l];
        for (int r = 0; r < 8; ++r) {
            int m = r + hi * 8;
            float v0 = c0[r] + bias;
            float v1 = c1[r] + bias;
            sF[m * 256 + col0 + l]        = v0 > 0.0f ? v0 : 0.0f;
            sF[(16 + m) * 256 + col0 + l] = v1 > 0.0f ? v1 : 0.0f;
        }
    }
    __syncthreads();

    // phase 2: 8 column-tiles of output; each is a 32x16 strip
    {
        int col0 = wave * 16;
        v8f c0 = {}, c1 = {};
        for (int k0 = 0; k0 < 256; k0 += 4) {
            v2f b;
            b.x = W2[(size_t)(k0 + 2 * hi) * 128 + col0 + l];
            b.y = W2[(size_t)(k0 + 2 * hi + 1) * 128 + col0 + l];
            v2f a0 = *(const v2f*)(sF + l * 256 + k0 + 2 * hi);
            v2f a1 = *(const v2f*)(sF + (16 + l) * 256 + k0 + 2 * hi);
            c0 = __builtin_amdgcn_wmma_f32_16x16x4_f32(false, a0, false, b,
                                                       (short)0, c0, false, false);
            c1 = __builtin_amdgcn_wmma_f32_16x16x4_f32(false, a1, false, b,
                                                       (short)0, c1, false, false);
        }
        float bias = b2[col0 + l];
        for (int r = 0; r < 8; ++r) {
            int m = r + hi * 8;
            float v0 = c0[r] + bias + sA[m * 128 + col0 + l];          // residual
            float v1 = c1[r] + bias + sA[(16 + m) * 128 + col0 + l];
            out[(size_t)(row0 + m) * 128 + col0 + l]      = v0;
            out[(size_t)(row0 + 16 + m) * 128 + col0 + l] = v1;
        }
    }
}

// ---------------------------------------------------------------------------
extern "C" void kernel_launch(void* const* d_in, const int* in_sizes, int n_in,
                              void* d_out, int out_size, void* d_ws, size_t ws_size,
                              hipStream_t stream) {
    const float* x      = (const float*)d_in[0];
    const int*   ei     = (const int*)  d_in[1];
    const float* W      = (const float*)d_in[2];
    const float* b      = (const float*)d_in[3];
    const float* gamma1 = (const float*)d_in[4];
    const float* beta1  = (const float*)d_in[5];
    const float* gamma2 = (const float*)d_in[6];
    const float* beta2  = (const float*)d_in[7];
    const float* gamma3 = (const float*)d_in[8];
    const float* beta3  = (const float*)d_in[9];
    const float* W1     = (const float*)d_in[10];
    const float* b1     = (const float*)d_in[11];
    const float* W2     = (const float*)d_in[12];
    const float* b2     = (const float*)d_in[13];
    float* outp = (float*)d_out;

    const int N = in_sizes[0] / D;
    const int E = in_sizes[1] / 2;
    const size_t ND = (size_t)N * D;
    const float invN = 1.0f / (float)N;

    float* buf0  = (float*)d_ws;          // hW -> h2pre
    float* buf1  = buf0 + ND;             // agg -> h3pre
    float* deg   = buf1 + ND;             // N
    float* sum   = deg + N;               // 128
    float* sumsq = sum + D;               // 128
    float* scale = sumsq + D;             // 128
    float* shift = scale + D;             // 128

    const int T256 = 256;
    int gN  = (N + T256 - 1) / T256;
    int gE  = (E + T256 - 1) / T256;
    int gEl = (int)((((long long)E + N) * WAVE + T256 - 1) / T256);
    int gTiles = ((N >> 5) * (D >> 4) + 7) / 8;   // 32x16 strip per wave
    int gElem  = 4096;
    int gStats = 960;

    // 1) degree + dinv
    deg_init_kernel<<<gN, T256, 0, stream>>>(deg, N);
    deg_acc_kernel<<<gE, T256, 0, stream>>>(ei + E, deg, E);
    deg_inv_kernel<<<gN, T256, 0, stream>>>(deg, N);

    // 2) hW = x @ W   (WMMA fp32)
    gemm_f32_kernel<<<gTiles, T256, 0, stream>>>(x, W, buf0, N, D, D);

    // 3) agg = b ; scatter-add normalized messages
    agg_init_kernel<<<gElem, T256, 0, stream>>>(buf1, b, ND);
    edge_scatter_kernel<<<gEl, T256, 0, stream>>>(ei, buf0, deg, buf1, E, N);

    // 4) bn1 + relu + residual -> h2pre (buf0)
    stats_zero_kernel<<<1, 256, 0, stream>>>(sum);
    colstats_kernel<<<gStats, T256, 0, stream>>>(buf1, sum, sumsq, N);
    bn_finalize_kernel<<<1, D, 0, stream>>>(sum, sumsq, gamma1, beta1, scale, shift, invN);
    bn_apply_kernel<<<gElem, T256, 0, stream>>>(buf1, x, scale, shift, buf0, ND, 1);

    // 5) bn2 -> h2 (d_out used as storage)
    stats_zero_kernel<<<1, 256, 0, stream>>>(sum);
    colstats_kernel<<<gStats, T256, 0, stream>>>(buf0, sum, sumsq, N);
    bn_finalize_kernel<<<1, D, 0, stream>>>(sum, sumsq, gamma2, beta2, scale, shift, invN);
    bn_apply_kernel<<<gElem, T256, 0, stream>>>(buf0, nullptr, scale, shift, outp, ND, 0);

    // 6) fused FFN + residual -> h3pre (buf1)
    ffn_fused_kernel<<<N / 32, T256, 0, stream>>>(outp, W1, b1, W2, b2, buf1, N);

    // 7) bn3 -> final output
    stats_zero_kernel<<<1, 256, 0, stream>>>(sum);
    colstats_kernel<<<gStats, T256, 0, stream>>>(buf1, sum, sumsq, N);
    bn_finalize_kernel<<<1, D, 0, stream>>>(sum, sumsq, gamma3, beta3, scale, shift, invN);
    bn_apply_kernel<<<gElem, T256, 0, stream>>>(buf1, nullptr, scale, shift, outp, ND, 0);
}